// TokenChoiceTopKRouter_52802327937613
// MI455X (gfx1250) — compile-verified
//
#include <hip/hip_runtime.h>
#include <math.h>

#define NUM_TOKENS 16384
#define DIM        4096
#define NEXP       8
#define TOPK       2

typedef __attribute__((ext_vector_type(2))) float v2f;
typedef __attribute__((ext_vector_type(4))) float v4f;
typedef __attribute__((ext_vector_type(8))) float v8f;

// ---------------------------------------------------------------------------
// Kernel 1: gate GEMM (x @ W^T) via V_WMMA_F32_16X16X4_F32 + sigmoid + top-2.
// One block = 16 tokens. 8 waves split K=4096 into 512-wide slices; partial
// 16x8 logit tiles are reduced through LDS.
//
// WMMA f32 16x16x4 operand layout (ISA 7.12.2):
//   A (16x4): lane L<16 -> row M=L, v0/v1 = K-slot 0/1 ; lane L>=16 -> slots 2/3
//   B (4x16): lane L<16 -> col N=L, v0/v1 = K-slot 0/1 ; lane L>=16 -> slots 2/3
//   C (16x16): lane L<16 -> N=L, VGPR v -> M=v ; lane L>=16 -> N=L-16, M=v+8
//
// K-chunk remap (valid because dot-product terms commute): per K=8 chunk each
// lane does ONE b128 load at (k + 4*hi); WMMA#1 consumes elements [0],[1],
// WMMA#2 consumes [2],[3]. A and B use the identical remap -> exact result.
//
// Lanes with n>=8 load W row (n&7) unconditionally: column n of D depends only
// on column n of B, and we discard D columns 8..15, so no masking is needed.
//
// x is a 256 MiB zero-reuse stream -> non-temporal loads (TH=NT) so it does
// not cycle the 192 MB L2; W (128 KiB, reused by all blocks) stays default RT.
// ---------------------------------------------------------------------------
__global__ __launch_bounds__(256)
void router_gate_topk(const float* __restrict__ x, const float* __restrict__ W,
                      float* __restrict__ topScores, int* __restrict__ topExperts) {
    __shared__ float partial[8][16][NEXP];   // per-wave partial logit tiles
    __shared__ float logits[16][NEXP];       // reduced logits

    const int tid  = threadIdx.x;
    const int wave = tid >> 5;
    const int lane = tid & 31;
    const int tokenBase = blockIdx.x * 16;

    const int m  = lane & 15;   // token row within tile (A side)
    const int n  = lane & 15;   // expert column (B side)
    const int hi = lane >> 4;   // K-phase select

    const float* __restrict__ xk = x + (size_t)(tokenBase + m) * DIM + 4 * hi;
    const float* __restrict__ wk = W + (size_t)(n & 7) * DIM + 4 * hi;

    v8f c = {};
    const int kBeg = wave * (DIM / 8);
    const int kEnd = kBeg + (DIM / 8);
#pragma unroll 4
    for (int k = kBeg; k < kEnd; k += 8) {
        const v4f av = __builtin_nontemporal_load((const v4f*)(xk + k)); // b128, TH=NT
        const v4f bv = *(const v4f*)(wk + k);                            // b128, RT
        v2f a0, b0, a1, b1;
        a0.x = av.x; a0.y = av.y;  b0.x = bv.x; b0.y = bv.y;
        a1.x = av.z; a1.y = av.w;  b1.x = bv.z; b1.y = bv.w;
        c = __builtin_amdgcn_wmma_f32_16x16x4_f32(false, a0, false, b0,
                                                  (short)0, c, false, false);
        c = __builtin_amdgcn_wmma_f32_16x16x4_f32(false, a1, false, b1,
                                                  (short)0, c, false, false);
    }

    if (n < NEXP) {
#pragma unroll
        for (int v = 0; v < 8; ++v)
            partial[wave][v + 8 * hi][n] = c[v];
    }
    __syncthreads();

    // Reduce the 8 K-slice partials: one thread per (token, expert) cell.
    if (tid < 16 * NEXP) {
        const int mm = tid >> 3, ee = tid & 7;
        float s = 0.0f;
#pragma unroll
        for (int w = 0; w < 8; ++w) s += partial[w][mm][ee];
        logits[mm][ee] = s;
    }
    __syncthreads();

    // Sigmoid + top-2 (descending; ties -> lowest expert index, like lax.top_k)
    if (tid < 16) {
        float sc[NEXP];
#pragma unroll
        for (int e = 0; e < NEXP; ++e)
            sc[e] = 1.0f / (1.0f + expf(-logits[tid][e]));

        int   b0 = 0;  float v0 = sc[0];
#pragma unroll
        for (int e = 1; e < NEXP; ++e)
            if (sc[e] > v0) { v0 = sc[e]; b0 = e; }
        int   b1 = 0;  float v1 = -1.0f;
#pragma unroll
        for (int e = 0; e < NEXP; ++e)
            if (e != b0 && sc[e] > v1) { v1 = sc[e]; b1 = e; }

        const int t = tokenBase + tid;
        topScores [t * TOPK + 0] = v0;
        topScores [t * TOPK + 1] = v1;
        topExperts[t * TOPK + 0] = b0;
        topExperts[t * TOPK + 1] = b1;
    }
}

// ---------------------------------------------------------------------------
// Kernel 2: deterministic stable counting sort of 32768 (token,expert) pairs
// by expert id (8 bins). Single block, 1024 threads, 32 entries each.
// out[0:32768)        = top_scores_sorted
// out[32768:65536)    = token_indices  (order // 2, as float)
// out[65536:65544)    = num_tokens_per_expert (as float)
// ---------------------------------------------------------------------------
__global__ __launch_bounds__(1024)
void router_sort_scatter(const float* __restrict__ topScores,
                         const int* __restrict__ topExperts,
                         float* __restrict__ out) {
    constexpr int N     = NUM_TOKENS * TOPK;  // 32768
    constexpr int NT    = 1024;
    constexpr int ITEMS = N / NT;             // 32

    __shared__ unsigned short scan[NT][NEXP]; // 16 KB, per-expert thread scan
    __shared__ int expertBase[NEXP];

    const int t    = threadIdx.x;
    const int base = t * ITEMS;

    unsigned short local[NEXP];
    int experts[ITEMS];
#pragma unroll
    for (int e = 0; e < NEXP; ++e) local[e] = 0;
#pragma unroll
    for (int i = 0; i < ITEMS; ++i) {
        const int e = topExperts[base + i];
        experts[i] = e;
        local[e]++;
    }
#pragma unroll
    for (int e = 0; e < NEXP; ++e) scan[t][e] = local[e];
    __syncthreads();

    // Hillis-Steele inclusive scan over threads, 8 experts in parallel.
    for (int s = 1; s < NT; s <<= 1) {
        unsigned short v[NEXP];
#pragma unroll
        for (int e = 0; e < NEXP; ++e)
            v[e] = (unsigned short)(scan[t][e] + ((t >= s) ? scan[t - s][e] : 0));
        __syncthreads();
#pragma unroll
        for (int e = 0; e < NEXP; ++e) scan[t][e] = v[e];
        __syncthreads();
    }

    if (t == 0) {
        int run = 0;
#pragma unroll
        for (int e = 0; e < NEXP; ++e) {
            expertBase[e] = run;
            const int tot = scan[NT - 1][e];
            out[2 * N + e] = (float)tot;       // histogram
            run += tot;
        }
    }
    __syncthreads();

    int pref[NEXP], run[NEXP];
#pragma unroll
    for (int e = 0; e < NEXP; ++e) {
        pref[e] = expertBase[e] + (int)scan[t][e] - (int)local[e]; // exclusive
        run[e]  = 0;
    }
#pragma unroll
    for (int i = 0; i < ITEMS; ++i) {
        const int e   = experts[i];
        const int pos = pref[e] + run[e]++;
        out[pos]     = topScores[base + i];
        out[N + pos] = (float)((base + i) >> 1); // token index = order // TOP_K
    }
}

extern "C" void kernel_launch(void* const* d_in, const int* in_sizes, int n_in,
                              void* d_out, int out_size, void* d_ws, size_t ws_size,
                              hipStream_t stream) {
    const float* x = (const float*)d_in[0];
    const float* W = (const float*)d_in[1];
    float* out = (float*)d_out;

    float* topScores  = (float*)d_ws;
    int*   topExperts = (int*)((char*)d_ws + (size_t)NUM_TOKENS * TOPK * sizeof(float));

    router_gate_topk<<<NUM_TOKENS / 16, 256, 0, stream>>>(x, W, topScores, topExperts);
    router_sort_scatter<<<1, 1024, 0, stream>>>(topScores, topExperts, out);
}